// NormalGAT_38714835206791
// MI455X (gfx1250) — compile-verified
//
#include <hip/hip_runtime.h>
#include <hip/hip_bf16.h>

typedef __attribute__((ext_vector_type(2))) float v2f;
typedef __attribute__((ext_vector_type(8))) float v8f;

#define GAT_B 4
#define GAT_T 12
#define GAT_N 2000
#define GAT_F 64
#define GAT_BT (GAT_B * GAT_T)          // 48
#define GAT_NE (GAT_BT * GAT_F)         // 3072 per node
#define GAT_TPB 256
#define GAT_PER (GAT_NE / GAT_TPB)      // 12 channels per thread
#define GAT_MAXDEG 512
#define NEG_SLOPE 0.01f

// ---------------------------------------------------------------------------
// Kernel 1: node projection GEMM with fp32 WMMA.
//   P[m, 0:128] = A[m, 0:64] @ Bm[0:64, 0:128]
//   A  = state viewed as [M=B*T*N, 64] (contiguous rows)
//   Bm[k, j]      = W[j, k]        (j <  64)  -> P1 = s @ W1^T
//   Bm[k, j]      = W[j-64, 64+k]  (j >= 64)  -> P2 = s @ W2^T
// One block = 8 waves; each wave owns one 16-wide N tile, block owns 16 M rows.
// ---------------------------------------------------------------------------
__global__ void __launch_bounds__(256)
gat_proj_wmma(const float* __restrict__ A, const float* __restrict__ W,
              float* __restrict__ P)
{
    const int lane = threadIdx.x & 31;
    const int wv   = threadIdx.x >> 5;        // 0..7 -> N tile index
    const int mrow = lane & 15;               // row within 16x16 tile
    const int half = lane >> 4;               // 0: K pair 0-1, 1: K pair 2-3
    const int Mbase = blockIdx.x * 16;
    const int Nbase = wv * 16;

    // Output column this lane covers (same for B operand and D result).
    const int j    = Nbase + mrow;
    const int col  = (j < 64) ? j : (j - 64);
    const int kofs = (j < 64) ? 0 : 64;

    const float* arow = A + (size_t)(Mbase + mrow) * 64 + 2 * half;
    const float* wcol = W + (size_t)col * 128 + kofs + 2 * half;

    v8f c = {};
#pragma unroll
    for (int kk = 0; kk < 64; kk += 4) {
        // A 16x4 fp32 layout: lanes 0-15 hold K=kk+0..1, lanes 16-31 K=kk+2..3
        v2f a = { arow[kk], arow[kk + 1] };
        // B 4x16 fp32 layout mirrors A: lanes 0-15 K rows kk+0..1, 16-31 kk+2..3
        v2f b = { wcol[kk], wcol[kk + 1] };
        c = __builtin_amdgcn_wmma_f32_16x16x4_f32(
                /*neg_a=*/false, a, /*neg_b=*/false, b,
                /*c_mod=*/(short)0, c, /*reuse_a=*/false, /*reuse_b=*/false);
    }

    // D layout: VGPR r -> M=r (lanes 0-15), M=8+r (lanes 16-31); N = lane&15.
    float* prow = P + (size_t)(Mbase + 8 * half) * 128 + (size_t)j;
#pragma unroll
    for (int r = 0; r < 8; ++r)
        prow[(size_t)r * 128] = c[r];
}

// ---------------------------------------------------------------------------
// Kernel 2: deterministic per-destination edge lists.
// One thread per node scans all edges in index order (dst[e] loads are
// wave-uniform -> broadcast). Edge order within a node is ascending e, so
// later FP reductions are bitwise deterministic.
// ---------------------------------------------------------------------------
__global__ void gat_build_edges(const int* __restrict__ dst,
                                int* __restrict__ deg,
                                int* __restrict__ eids, int E)
{
    const int n = blockIdx.x * blockDim.x + threadIdx.x;
    if (n >= GAT_N) return;
    int cnt = 0;
    for (int e = 0; e < E; ++e) {
        if (dst[e] == n) {
            if (cnt < GAT_MAXDEG) eids[cnt * GAT_N + n] = e;
            ++cnt;
        }
    }
    deg[n] = (cnt < GAT_MAXDEG) ? cnt : GAT_MAXDEG;
}

// ---------------------------------------------------------------------------
// Kernel 3: fused edge phase. One block per destination node n.
//   alpha[e] = leaky(P1[src[e]] + (P2[n] + bias)) * dist[e]
//   segment softmax over incoming edges, aggregate ex * s_src, ReLU.
// P2[n]+bias cached in LDS (12 KB); each thread owns 12 (bt,f) channels in
// registers for max/denom/numerator. Two passes over the (small, L2-resident)
// edge list; P rows hit L2.
// ---------------------------------------------------------------------------
__global__ void __launch_bounds__(256)
gat_edge_softmax(const float* __restrict__ state, const float* __restrict__ P,
                 const float* __restrict__ bias, const int* __restrict__ src,
                 const float* __restrict__ dist, const int* __restrict__ deg,
                 const int* __restrict__ eids, float* __restrict__ out)
{
    __shared__ float p2b[GAT_NE];

    const int n   = blockIdx.x;
    const int tid = threadIdx.x;

    for (int i = tid; i < GAT_NE; i += GAT_TPB) {
        const int bt = i >> 6, f = i & 63;
        p2b[i] = P[((size_t)(bt * GAT_N + n)) * 128 + 64 + f] + bias[f];
    }
    __syncthreads();

    const int d = deg[n];

    float mx[GAT_PER];
#pragma unroll
    for (int j = 0; j < GAT_PER; ++j) mx[j] = -3.0e38f;

    // Pass 1: per-channel max of alpha over incoming edges.
    for (int k = 0; k < d; ++k) {
        const int   e  = eids[k * GAT_N + n];
        const int   s  = src[e];
        const float de = dist[e];
#pragma unroll
        for (int j = 0; j < GAT_PER; ++j) {
            const int i  = tid + j * GAT_TPB;
            const int bt = i >> 6, f = i & 63;
            float a = P[((size_t)(bt * GAT_N + s)) * 128 + f] + p2b[i];
            a = (a >= 0.f) ? a : NEG_SLOPE * a;
            a *= de;
            mx[j] = fmaxf(mx[j], a);
        }
    }

    float den[GAT_PER], num[GAT_PER];
#pragma unroll
    for (int j = 0; j < GAT_PER; ++j) { den[j] = 0.f; num[j] = 0.f; }

    // Pass 2: exp-sum denominator and ex-weighted source aggregation.
    for (int k = 0; k < d; ++k) {
        const int   e  = eids[k * GAT_N + n];
        const int   s  = src[e];
        const float de = dist[e];
#pragma unroll
        for (int j = 0; j < GAT_PER; ++j) {
            const int i  = tid + j * GAT_TPB;
            const int bt = i >> 6, f = i & 63;
            float a = P[((size_t)(bt * GAT_N + s)) * 128 + f] + p2b[i];
            a = (a >= 0.f) ? a : NEG_SLOPE * a;
            a *= de;
            const float ex = __expf(a - mx[j]);
            den[j] += ex;
            num[j] += ex * state[((size_t)(bt * GAT_N + s)) * 64 + f];
        }
    }

    // new_state = relu( sum(ex * s_src) / sum(ex) ); nodes with no edges -> 0.
#pragma unroll
    for (int j = 0; j < GAT_PER; ++j) {
        const int i  = tid + j * GAT_TPB;
        const int bt = i >> 6, f = i & 63;
        const float v = (den[j] > 0.f) ? (num[j] / den[j]) : 0.f;
        out[((size_t)(bt * GAT_N + n)) * 64 + f] = fmaxf(v, 0.f);
    }
}

// ---------------------------------------------------------------------------
extern "C" void kernel_launch(void* const* d_in, const int* in_sizes, int n_in,
                              void* d_out, int out_size, void* d_ws, size_t ws_size,
                              hipStream_t stream)
{
    const float* state = (const float*)d_in[0];
    // d_in[1] = feature: unused by the reference computation.
    const float* W     = (const float*)d_in[2];   // [64, 128] row-major
    const float* bias  = (const float*)d_in[3];   // [64]
    const int*   src   = (const int*)d_in[4];     // [E]
    const int*   dst   = (const int*)d_in[5];     // [E]
    const float* dist  = (const float*)d_in[6];   // [E,1]
    float*       out   = (float*)d_out;           // [B,T,N,F]

    const int E = in_sizes[4];
    const int M = GAT_B * GAT_T * GAT_N;          // 96000 GEMM rows

    // Workspace layout: P (M*128 f32, 49.15 MB) | deg (N i32) | eids (N*MAXDEG i32)
    char*  ws   = (char*)d_ws;
    float* P    = (float*)ws;
    int*   deg  = (int*)(ws + (size_t)M * 128 * sizeof(float));
    int*   eids = deg + GAT_N;

    gat_proj_wmma<<<M / 16, 256, 0, stream>>>(state, W, P);
    gat_build_edges<<<(GAT_N + 255) / 256, 256, 0, stream>>>(dst, deg, eids, E);
    gat_edge_softmax<<<GAT_N, GAT_TPB, 0, stream>>>(state, P, bias, src, dist,
                                                    deg, eids, out);
}